// EdgeConv2d_75179107549327
// MI455X (gfx1250) — compile-verified
//
#include <hip/hip_runtime.h>

typedef __attribute__((ext_vector_type(2))) float v2f;
typedef __attribute__((ext_vector_type(8))) float v8f;

// Problem sizes (fixed by reference)
#define Bsz   8
#define Csz   64
#define Nsz   4096
#define Ksz   16
#define OUTsz 64

// ---------------------------------------------------------------------------
// Prep 1: aw[0][m][c] = W[m][c] - W[m][64+c]   (= W1 - W2)
//         aw[1][m][c] = W[m][64+c]             (= W2)
// 4096 threads total.
// ---------------------------------------------------------------------------
__global__ __launch_bounds__(256) void prep_aw(const float* __restrict__ w,
                                               float* __restrict__ aw) {
    int t = blockIdx.x * 256 + threadIdx.x;      // [0, 4096)
    int m = t >> 6;
    int c = t & 63;
    float w1 = w[m * 128 + c];
    float w2 = w[m * 128 + 64 + c];
    aw[t]        = w1 - w2;   // mat 0
    aw[4096 + t] = w2;        // mat 1
}

// ---------------------------------------------------------------------------
// Prep 2: transpose x (B,C,N) -> xt (B,N,C) so WMMA B-operand pairs and the
// later per-node gathers are contiguous.
// ---------------------------------------------------------------------------
__global__ __launch_bounds__(256) void prep_xt(const float* __restrict__ x,
                                               float* __restrict__ xt) {
    int t = blockIdx.x * 256 + threadIdx.x;      // [0, B*N*C) = [0, 2M)
    int b = t >> 18;                             // N*C = 262144
    int r = t & 262143;
    int n = r >> 6;
    int c = r & 63;
    xt[t] = x[(((b << 6) + c) << 12) + n];
}

// ---------------------------------------------------------------------------
// GEMM: per batch b, U(64x4096) = (W1-W2) * Xb, V(64x4096) = W2 * Xb
// using V_WMMA_F32_16X16X4_F32 (full fp32). One wave per 16x16 output tile
// column set; A tiles live in registers for the whole wave, B loads are
// shared between the U and V accumulations.
// Grid: 512 blocks * 128 threads (4 waves). blk -> (b, mtile, nchunk).
// ---------------------------------------------------------------------------
__global__ __launch_bounds__(128) void gemm_uv(const float* __restrict__ aw,
                                               const float* __restrict__ xt,
                                               float* __restrict__ U,
                                               float* __restrict__ V) {
    const int lane  = threadIdx.x & 31;
    const int wave  = threadIdx.x >> 5;
    const int g     = lane >> 4;     // half-wave select
    const int lm    = lane & 15;
    const int blk   = blockIdx.x;
    const int b     = blk >> 6;      // 64 blocks per batch
    const int rem   = blk & 63;
    const int mtile = rem >> 4;      // 4 M-tiles of 16
    const int nchunk= rem & 15;      // 16 chunks of 16 ntiles
    const int mrow  = mtile * 16 + lm;

    // A-matrix 16x4 fp32 layout: lane(g,m): VGPR0=A[m][2g], VGPR1=A[m][2g+1]
    v2f a0[16], a1[16];
#pragma unroll
    for (int ks = 0; ks < 16; ++ks) {
        int kk = ks * 4 + 2 * g;
        a0[ks] = *(const v2f*)(aw +        (mrow << 6) + kk);
        a1[ks] = *(const v2f*)(aw + 4096 + (mrow << 6) + kk);
    }

    const float* xb = xt + ((size_t)b << 18);    // b * N * C

#pragma unroll
    for (int it = 0; it < 4; ++it) {
        int ntile = nchunk * 16 + wave * 4 + it; // [0,256)
        int n     = ntile * 16 + lm;             // this lane's column
        const float* xcol = xb + ((size_t)n << 6);

        v8f accU = {};
        v8f accV = {};
#pragma unroll
        for (int ks = 0; ks < 16; ++ks) {
            // B-matrix 4x16 fp32 layout: lane(g,n): VGPR0=X[2g][n], VGPR1=X[2g+1][n]
            v2f bb = *(const v2f*)(xcol + ks * 4 + 2 * g);
            accU = __builtin_amdgcn_wmma_f32_16x16x4_f32(
                false, a0[ks], false, bb, (short)0, accU, false, false);
            accV = __builtin_amdgcn_wmma_f32_16x16x4_f32(
                false, a1[ks], false, bb, (short)0, accV, false, false);
        }

        // D layout: lanes 0-15 -> (M=r, N=lane), lanes 16-31 -> (M=r+8, N=lane-16)
        // Store into (b, n, c)-major so gathers later read contiguous rows.
        size_t base = (((size_t)b << 12) + (size_t)n) * 64 + (size_t)(mtile * 16 + 8 * g);
        *(v8f*)(U + base) = accU;   // 32B aligned (base multiple of 8)
        *(v8f*)(V + base) = accV;
    }
}

// ---------------------------------------------------------------------------
// Gather + bias + relu + max over K.
// out[b,o,n] = max_k relu( U[b, i(b,n,k), o] + V[b, j(b,n,k), o] + bias[o] )
// 16-lane group per node n; each lane covers 4 channels (float4).
// Grid: B * (N/16) = 2048 blocks * 256 threads.
// ---------------------------------------------------------------------------
__global__ __launch_bounds__(256) void edge_max(const int* __restrict__ ei,
                                                const float* __restrict__ U,
                                                const float* __restrict__ V,
                                                const float* __restrict__ bias,
                                                float* __restrict__ out) {
    const int b    = blockIdx.x >> 8;
    const int nblk = blockIdx.x & 255;
    const int n    = nblk * 16 + (threadIdx.x >> 4);
    const int l    = threadIdx.x & 15;           // channel group, o = 4l..4l+3

    const float4 bias4 = ((const float4*)bias)[l];

    // edge_index flat layout (2, B, N, K): d=0 -> j (source), d=1 -> i (target)
    const size_t eBase_j = (((size_t)b << 12) + (size_t)n) << 4;
    const size_t eBase_i = ((size_t)Bsz * Nsz * Ksz) + eBase_j;

    const float4* Ub = (const float4*)(U + ((size_t)b << 18));
    const float4* Vb = (const float4*)(V + ((size_t)b << 18));

    int idxi[Ksz], idxj[Ksz];
#pragma unroll
    for (int k = 0; k < Ksz; ++k) {
        idxi[k] = ei[eBase_i + k];
        idxj[k] = ei[eBase_j + k];
    }

    float bx = 0.f, by = 0.f, bz = 0.f, bw = 0.f;  // relu output >= 0
#pragma unroll
    for (int k = 0; k < Ksz; ++k) {
        float4 u = Ub[((size_t)idxi[k] << 4) + l];
        float4 v = Vb[((size_t)idxj[k] << 4) + l];
        float yx = fmaxf(u.x + v.x + bias4.x, 0.f);
        float yy = fmaxf(u.y + v.y + bias4.y, 0.f);
        float yz = fmaxf(u.z + v.z + bias4.z, 0.f);
        float yw = fmaxf(u.w + v.w + bias4.w, 0.f);
        bx = fmaxf(bx, yx);
        by = fmaxf(by, yy);
        bz = fmaxf(bz, yz);
        bw = fmaxf(bw, yw);
    }

    // out layout (B, OUT, N)
    size_t ob = (((size_t)b << 6) + (size_t)(4 * l)) * (size_t)Nsz + (size_t)n;
    out[ob]            = bx;
    out[ob + Nsz]      = by;
    out[ob + 2 * Nsz]  = bz;
    out[ob + 3 * Nsz]  = bw;
}

// ---------------------------------------------------------------------------
extern "C" void kernel_launch(void* const* d_in, const int* in_sizes, int n_in,
                              void* d_out, int out_size, void* d_ws, size_t ws_size,
                              hipStream_t stream) {
    const float* x    = (const float*)d_in[0];   // (B, C, N, 1)
    const int*   ei   = (const int*)  d_in[1];   // (2, B, N, K)
    const float* w    = (const float*)d_in[2];   // (OUT, 2C)
    const float* bias = (const float*)d_in[3];   // (OUT,)
    float* out = (float*)d_out;                  // (B, OUT, N)

    // Workspace carve-up (floats):
    //   aw : 2*64*64          =     8192
    //   xt : B*N*C            =  2097152
    //   U  : B*N*OUT          =  2097152
    //   V  : B*N*OUT          =  2097152
    float* aw = (float*)d_ws;
    float* xt = aw + 8192;
    float* U  = xt + (size_t)Bsz * Nsz * Csz;
    float* V  = U  + (size_t)Bsz * Nsz * OUTsz;

    prep_aw<<<16,   256, 0, stream>>>(w, aw);
    prep_xt<<<8192, 256, 0, stream>>>(x, xt);
    gemm_uv<<<512,  128, 0, stream>>>(aw, xt, U, V);
    edge_max<<<Bsz * (Nsz / 16), 256, 0, stream>>>(ei, U, V, bias, out);

    (void)in_sizes; (void)n_in; (void)out_size; (void)ws_size;
}